// GATEncoder_46205258170494
// MI455X (gfx1250) — compile-verified
//
#include <hip/hip_runtime.h>

// GAT encoder, 2 layers. N=100000, E=1.7M (incl self loops), F_IN=128, H=8, D=8.
// d_in: 0:x 1:edge_index(i64) 2:W1 3:a_src1 4:a_dst1 5:b1 6:W2 7:a_src2 8:a_dst2 9:b2

#define FIN   128
#define FMID  64      // HEADS * F_HID
#define NHEAD 8
#define SLOPE 0.2f

typedef __attribute__((ext_vector_type(2))) float v2f;
typedef __attribute__((ext_vector_type(8))) float v8f;

__device__ __forceinline__ float lrelu(float x) { return x > 0.f ? x : SLOPE * x; }

// float atomic-max via signed/unsigned integer monotonicity (init must be -inf)
__device__ __forceinline__ void atomicMaxF(float* a, float v) {
    if (v >= 0.f) atomicMax((int*)a, __float_as_int(v));
    else          atomicMin((unsigned int*)a, __float_as_uint(v));
}

// ---------------------------------------------------------------------------
// fp32 GEMM via V_WMMA_F32_16X16X4_F32. H[n x NO] = X[n x KD] * W[KD x NO].
// One wave computes a 16-row M tile across all NO columns (NO/16 acc tiles).
// A frag (16x4): lane half hl holds K = k0+2*hl (+1) -> contiguous float2.
// B frag (4x16): b.x = W[k0+2*hl][n16], b.y = W[k0+2*hl+1][n16].
// C/D: acc[t][v] is element (M = v + 8*hl, N = lane&15) of tile t.
// ---------------------------------------------------------------------------
template <int KD, int NO>
__global__ __launch_bounds__(256) void gemm_wmma_f32(const float* __restrict__ X,
                                                     const float* __restrict__ W,
                                                     float* __restrict__ H, int n) {
    constexpr int NT = NO / 16;
    const int wv    = threadIdx.x >> 5;   // wave in block: 0..7
    const int lane  = threadIdx.x & 31;
    const int hl    = lane >> 4;          // half-wave
    const int l16   = lane & 15;
    const int mbase = (blockIdx.x * 8 + wv) * 16;
    if (mbase >= n) return;               // wave-uniform: EXEC stays all-ones
    int row = mbase + l16;
    if (row >= n) row = n - 1;            // clamp loads for tail tile
    const float* xr = X + (size_t)row * KD;

    v8f acc[NT] = {};
    for (int k0 = 0; k0 < KD; k0 += 4) {
        const int ka = k0 + 2 * hl;
        v2f a; a.x = xr[ka]; a.y = xr[ka + 1];
        const float* w0 = W + (size_t)ka * NO + l16;
#pragma unroll
        for (int t = 0; t < NT; ++t) {
            v2f b; b.x = w0[16 * t]; b.y = w0[16 * t + NO];
            acc[t] = __builtin_amdgcn_wmma_f32_16x16x4_f32(
                false, a, false, b, (short)0, acc[t], false, false);
        }
    }
#pragma unroll
    for (int v = 0; v < 8; ++v) {
        const int m = mbase + v + 8 * hl;
        if (m < n) {
            float* o = H + (size_t)m * NO + l16;
#pragma unroll
            for (int t = 0; t < NT; ++t) o[16 * t] = acc[t][v];
        }
    }
}

// ---------------------------------------------------------------------------
// per-(node,head) attention terms: S[i,h] = <h1[i,h,:], a_src[h,:]>, likewise D
// ---------------------------------------------------------------------------
__global__ void attn_terms1_k(const float* __restrict__ H1, const float* __restrict__ AS,
                              const float* __restrict__ AD, float* __restrict__ S,
                              float* __restrict__ D, int n) {
    int t = blockIdx.x * blockDim.x + threadIdx.x;
    if (t >= n * NHEAD) return;
    int i = t >> 3, h = t & 7;
    const float4* hp = (const float4*)(H1 + (size_t)i * FMID + h * 8);
    const float4* as = (const float4*)(AS + h * 8);
    const float4* ad = (const float4*)(AD + h * 8);
    float4 h0 = hp[0], h1 = hp[1];
    float4 a0 = as[0], a1 = as[1];
    float4 c0 = ad[0], c1 = ad[1];
    S[t] = h0.x*a0.x + h0.y*a0.y + h0.z*a0.z + h0.w*a0.w
         + h1.x*a1.x + h1.y*a1.y + h1.z*a1.z + h1.w*a1.w;
    D[t] = h0.x*c0.x + h0.y*c0.y + h0.z*c0.z + h0.w*c0.w
         + h1.x*c1.x + h1.y*c1.y + h1.z*c1.z + h1.w*c1.w;
}

// per-node (single head) terms over 128 features
__global__ void attn_terms2_k(const float* __restrict__ H, const float* __restrict__ AS,
                              const float* __restrict__ AD, float* __restrict__ S,
                              float* __restrict__ D, int n) {
    int i = blockIdx.x * blockDim.x + threadIdx.x;
    if (i >= n) return;
    const float4* hp = (const float4*)(H + (size_t)i * FIN);
    float s = 0.f, d = 0.f;
#pragma unroll
    for (int q = 0; q < FIN / 4; ++q) {
        float4 h4 = hp[q];
        float4 a4 = ((const float4*)AS)[q];
        float4 c4 = ((const float4*)AD)[q];
        s += h4.x*a4.x + h4.y*a4.y + h4.z*a4.z + h4.w*a4.w;
        d += h4.x*c4.x + h4.y*c4.y + h4.z*c4.z + h4.w*c4.w;
    }
    S[i] = s; D[i] = d;
}

__global__ void fill_k(float* __restrict__ p, size_t n, float v) {
    size_t i = (size_t)blockIdx.x * blockDim.x + threadIdx.x;
    if (i < n) p[i] = v;
}

__global__ void bias_bcast_k(float* __restrict__ o, const float* __restrict__ b,
                             int mask, size_t n) {
    size_t i = (size_t)blockIdx.x * blockDim.x + threadIdx.x;
    if (i < n) o[i] = b[i & mask];
}

__global__ void bias_relu_k(float* __restrict__ o, const float* __restrict__ b,
                            int mask, size_t n) {
    size_t i = (size_t)blockIdx.x * blockDim.x + threadIdx.x;
    if (i < n) { float v = o[i] + b[i & mask]; o[i] = v > 0.f ? v : 0.f; }
}

// ------------------------- layer 1 edge passes (H=8) -----------------------
__global__ void edge_max1_k(const long long* __restrict__ src, const long long* __restrict__ dst,
                            const float* __restrict__ S, const float* __restrict__ D,
                            float* __restrict__ E1, float* __restrict__ M1, int nE) {
    int e = blockIdx.x * blockDim.x + threadIdx.x;
    if (e >= nE) return;
    int s = (int)src[e], d = (int)dst[e];
    const float4* sp = (const float4*)(S + (size_t)s * 8);
    const float4* dp = (const float4*)(D + (size_t)d * 8);
    float4 sa = sp[0], sb = sp[1], da = dp[0], db = dp[1];
    float4 ea, eb;
    ea.x = lrelu(sa.x + da.x); ea.y = lrelu(sa.y + da.y);
    ea.z = lrelu(sa.z + da.z); ea.w = lrelu(sa.w + da.w);
    eb.x = lrelu(sb.x + db.x); eb.y = lrelu(sb.y + db.y);
    eb.z = lrelu(sb.z + db.z); eb.w = lrelu(sb.w + db.w);
    ((float4*)(E1 + (size_t)e * 8))[0] = ea;
    ((float4*)(E1 + (size_t)e * 8))[1] = eb;
    float* m = M1 + (size_t)d * 8;
    atomicMaxF(m + 0, ea.x); atomicMaxF(m + 1, ea.y);
    atomicMaxF(m + 2, ea.z); atomicMaxF(m + 3, ea.w);
    atomicMaxF(m + 4, eb.x); atomicMaxF(m + 5, eb.y);
    atomicMaxF(m + 6, eb.z); atomicMaxF(m + 7, eb.w);
}

__global__ void edge_exp1_k(const long long* __restrict__ dst, float* __restrict__ E1,
                            const float* __restrict__ M1, float* __restrict__ DEN, int nE) {
    int e = blockIdx.x * blockDim.x + threadIdx.x;
    if (e >= nE) return;
    int d = (int)dst[e];
    const float* m = M1 + (size_t)d * 8;
    float* ep = E1 + (size_t)e * 8;
    float* dn = DEN + (size_t)d * 8;
#pragma unroll
    for (int h = 0; h < 8; ++h) {
        float ex = expf(ep[h] - m[h]);
        ep[h] = ex;
        atomicAdd(dn + h, ex);
    }
}

// one thread per (edge, head): scatter alpha-weighted 8-float message
__global__ void edge_agg1_k(const long long* __restrict__ src, const long long* __restrict__ dst,
                            const float* __restrict__ E1, const float* __restrict__ DEN,
                            const float* __restrict__ H1, float* __restrict__ OUT, int nEH) {
    int t = blockIdx.x * blockDim.x + threadIdx.x;
    if (t >= nEH) return;
    int e = t >> 3, h = t & 7;
    int s = (int)src[e], d = (int)dst[e];
    float alpha = E1[(size_t)e * 8 + h] / DEN[(size_t)d * 8 + h];
    const float4* hp = (const float4*)(H1 + (size_t)s * FMID + h * 8);
    float4 h0 = hp[0], h1 = hp[1];
    float* o = OUT + (size_t)d * FMID + h * 8;
    atomicAdd(o + 0, h0.x * alpha); atomicAdd(o + 1, h0.y * alpha);
    atomicAdd(o + 2, h0.z * alpha); atomicAdd(o + 3, h0.w * alpha);
    atomicAdd(o + 4, h1.x * alpha); atomicAdd(o + 5, h1.y * alpha);
    atomicAdd(o + 6, h1.z * alpha); atomicAdd(o + 7, h1.w * alpha);
}

// ------------------------- layer 2 edge passes (H=1) -----------------------
__global__ void edge_max2_k(const long long* __restrict__ src, const long long* __restrict__ dst,
                            const float* __restrict__ S, const float* __restrict__ D,
                            float* __restrict__ E2, float* __restrict__ M2, int nE) {
    int e = blockIdx.x * blockDim.x + threadIdx.x;
    if (e >= nE) return;
    int s = (int)src[e], d = (int)dst[e];
    float v = lrelu(S[s] + D[d]);
    E2[e] = v;
    atomicMaxF(M2 + d, v);
}

__global__ void edge_exp2_k(const long long* __restrict__ dst, float* __restrict__ E2,
                            const float* __restrict__ M2, float* __restrict__ DEN, int nE) {
    int e = blockIdx.x * blockDim.x + threadIdx.x;
    if (e >= nE) return;
    int d = (int)dst[e];
    float ex = expf(E2[e] - M2[d]);
    E2[e] = ex;
    atomicAdd(DEN + d, ex);
}

// one thread per (edge, 16-float chunk): 8 threads cover the 128-wide message
__global__ void edge_agg2_k(const long long* __restrict__ src, const long long* __restrict__ dst,
                            const float* __restrict__ E2, const float* __restrict__ DEN,
                            const float* __restrict__ HH, float* __restrict__ OUT, int nEC) {
    int t = blockIdx.x * blockDim.x + threadIdx.x;
    if (t >= nEC) return;
    int e = t >> 3, c = t & 7;
    int s = (int)src[e], d = (int)dst[e];
    float alpha = E2[e] / DEN[d];
    const float4* hp = (const float4*)(HH + (size_t)s * FIN + c * 16);
    float* o = OUT + (size_t)d * FIN + c * 16;
#pragma unroll
    for (int q = 0; q < 4; ++q) {
        float4 h4 = hp[q];
        atomicAdd(o + q * 4 + 0, h4.x * alpha);
        atomicAdd(o + q * 4 + 1, h4.y * alpha);
        atomicAdd(o + q * 4 + 2, h4.z * alpha);
        atomicAdd(o + q * 4 + 3, h4.w * alpha);
    }
}

static inline int gb(size_t n, int bs = 256) { return (int)((n + bs - 1) / bs); }

extern "C" void kernel_launch(void* const* d_in, const int* in_sizes, int n_in,
                              void* d_out, int out_size, void* d_ws, size_t ws_size,
                              hipStream_t stream) {
    const float*     x      = (const float*)d_in[0];
    const long long* ei     = (const long long*)d_in[1];
    const float*     W1     = (const float*)d_in[2];
    const float*     a_src1 = (const float*)d_in[3];
    const float*     a_dst1 = (const float*)d_in[4];
    const float*     b1     = (const float*)d_in[5];
    const float*     W2     = (const float*)d_in[6];
    const float*     a_src2 = (const float*)d_in[7];
    const float*     a_dst2 = (const float*)d_in[8];
    const float*     b2     = (const float*)d_in[9];
    float*           out    = (float*)d_out;

    const int n  = in_sizes[0] / FIN;   // 100000
    const int nE = in_sizes[1] / 2;     // 1700000
    const long long* src = ei;
    const long long* dst = ei + nE;

    // workspace carve-out (floats)
    float* ws = (float*)d_ws;
    size_t o = 0;
    float* h1   = ws + o; o += (size_t)n * FMID;   // layer-1 features [N,64]
    float* s1   = ws + o; o += (size_t)n * NHEAD;
    float* d1   = ws + o; o += (size_t)n * NHEAD;
    float* m1   = ws + o; o += (size_t)n * NHEAD;
    float* den1 = ws + o; o += (size_t)n * NHEAD;
    float* out1 = ws + o; o += (size_t)n * FMID;   // layer-1 aggregate -> layer-2 input
    float* e1   = ws + o; o += (size_t)nE * NHEAD; // per-edge logits / exp
    float* hh2  = ws + o; o += (size_t)n * FIN;    // layer-2 features [N,128]
    float* s2   = ws + o; o += (size_t)n;
    float* d2   = ws + o; o += (size_t)n;
    float* m2   = ws + o; o += (size_t)n;
    float* den2 = ws + o; o += (size_t)n;
    float* e2   = ws + o; o += (size_t)nE;

    const float ninf = -__builtin_inff();

    // ---- layer 1 ----
    gemm_wmma_f32<FIN, FMID><<<(n + 127) / 128, 256, 0, stream>>>(x, W1, h1, n);
    attn_terms1_k<<<gb((size_t)n * NHEAD), 256, 0, stream>>>(h1, a_src1, a_dst1, s1, d1, n);
    fill_k<<<gb((size_t)n * NHEAD), 256, 0, stream>>>(m1, (size_t)n * NHEAD, ninf);
    fill_k<<<gb((size_t)n * NHEAD), 256, 0, stream>>>(den1, (size_t)n * NHEAD, 0.f);
    fill_k<<<gb((size_t)n * FMID), 256, 0, stream>>>(out1, (size_t)n * FMID, 0.f);
    edge_max1_k<<<gb(nE), 256, 0, stream>>>(src, dst, s1, d1, e1, m1, nE);
    edge_exp1_k<<<gb(nE), 256, 0, stream>>>(dst, e1, m1, den1, nE);
    edge_agg1_k<<<gb((size_t)nE * NHEAD), 256, 0, stream>>>(src, dst, e1, den1, h1, out1,
                                                            nE * NHEAD);
    bias_relu_k<<<gb((size_t)n * FMID), 256, 0, stream>>>(out1, b1, FMID - 1, (size_t)n * FMID);

    // ---- layer 2 ----
    gemm_wmma_f32<FMID, FIN><<<(n + 127) / 128, 256, 0, stream>>>(out1, W2, hh2, n);
    attn_terms2_k<<<gb(n), 256, 0, stream>>>(hh2, a_src2, a_dst2, s2, d2, n);
    fill_k<<<gb(n), 256, 0, stream>>>(m2, (size_t)n, ninf);
    fill_k<<<gb(n), 256, 0, stream>>>(den2, (size_t)n, 0.f);
    bias_bcast_k<<<gb((size_t)n * FIN), 256, 0, stream>>>(out, b2, FIN - 1, (size_t)n * FIN);
    edge_max2_k<<<gb(nE), 256, 0, stream>>>(src, dst, s2, d2, e2, m2, nE);
    edge_exp2_k<<<gb(nE), 256, 0, stream>>>(dst, e2, m2, den2, nE);
    edge_agg2_k<<<gb((size_t)nE * 8), 256, 0, stream>>>(src, dst, e2, den2, hh2, out, nE * 8);
}